// AutoAttention_Layer_31782757991200
// MI455X (gfx1250) — compile-verified
//
#include <hip/hip_runtime.h>
#include <math.h>

// MI455X / gfx1250, wave32. One block per batch element.
// Memory-bound kernel: ~93 MB effective traffic -> ~4us at 23.3 TB/s.
// Masked column-sum of k done with V_WMMA_F32_16X16X4_F32 (exact f32 math).

typedef __attribute__((ext_vector_type(2))) float v2f;
typedef __attribute__((ext_vector_type(8))) float v8f;

__global__ __launch_bounds__(192)
void aa_kernel(const float* __restrict__ q,      // (B, 64, 64)
               const float* __restrict__ k,      // (B, 200, 192)
               const int*   __restrict__ kes,    // (B, 1)
               const float* __restrict__ mss,    // (3, 3)
               float*       __restrict__ out)    // (B, 1, 192)
{
    constexpr int Lq = 64, Lk = 200, D = 64, KC = 192;

    const int b    = blockIdx.x;
    const int t    = threadIdx.x;       // 0..191
    const int lane = t & 31;
    const int w    = t >> 5;            // wave id 0..5

    __shared__ float lds_q[D];          // qsum
    __shared__ float lds_ksum[KC];      // masked column sums of k
    __shared__ float lds_s[KC];         // pre-softmax logits
    __shared__ float lds_e[KC];         // exp values

    const int len = kes[b];

    // ---- qsum[d] = sum_j q[b,j,d] (threads 0..63, coalesced rows) ----
    if (t < D) {
        const float* qb = q + (size_t)b * Lq * D;
        float acc = 0.f;
        #pragma unroll 8
        for (int j = 0; j < Lq; ++j) acc += qb[j * D + t];
        lds_q[t] = acc;
    }

    // ---- masked column-sum of k via WMMA f32 16x16x4 ----
    // Wave w owns columns [32w, 32w+32): two 16-wide tiles.
    // A (16x4): A[m,kk] = mask[l0+kk], replicated over m  -> every row of D
    //           equals the masked column sum.
    // Operand layout (wave32): lane = M (A) / N (B); VGPR0 holds K={0 | 2}
    // for lane halves {0-15 | 16-31}, VGPR1 holds K={1 | 3}.
    const float* kb   = k + (size_t)b * Lk * KC;
    const int    half = lane >> 4;        // 0: K=0,1   1: K=2,3
    const int    n    = lane & 15;        // column within 16-wide tile
    const int    col0 = 32 * w;

    v8f c0 = {};
    v8f c1 = {};

    const int nch = (len + 3) >> 2;       // chunks of 4 rows; len <= 199 so
                                          // max row touched = 199 < Lk
    for (int ch = 0; ch < nch; ++ch) {
        const int l0 = ch * 4;
        const int ra = l0 + 2 * half;     // first K-row this lane-half holds

        v2f a;
        a.x = (ra     < len) ? 1.0f : 0.0f;   // K = 0 (or 2)
        a.y = (ra + 1 < len) ? 1.0f : 0.0f;   // K = 1 (or 3)

        const float* rp0 = kb + (size_t)ra * KC;
        const float* rp1 = rp0 + KC;

        v2f b0, b1;
        b0.x = rp0[col0 + n];          // tile 0, K-row ra
        b0.y = rp1[col0 + n];          // tile 0, K-row ra+1
        b1.x = rp0[col0 + 16 + n];     // tile 1
        b1.y = rp1[col0 + 16 + n];

        c0 = __builtin_amdgcn_wmma_f32_16x16x4_f32(
                 false, a, false, b0, (short)0, c0, false, false);
        c1 = __builtin_amdgcn_wmma_f32_16x16x4_f32(
                 false, a, false, b1, (short)0, c1, false, false);
    }

    // C/D layout: VGPR0, lanes 0-15 = row M=0, N=lane -> the column sum.
    if (lane < 16) {
        lds_ksum[col0 + lane]      = c0[0];
        lds_ksum[col0 + 16 + lane] = c1[0];
    }
    __syncthreads();

    // ---- epilogue: s[r,d] = qsum[d] * sum_s W[r,s]*ksum[s*64+d] ----
    const int r = t >> 6;                 // 0..2
    const int d = t & 63;                 // 0..63

    // W = softmax(mss_weight, axis=1), row r (9 broadcast scalars, cached)
    const float w0 = mss[r * 3 + 0];
    const float w1 = mss[r * 3 + 1];
    const float w2 = mss[r * 3 + 2];
    const float wm = fmaxf(w0, fmaxf(w1, w2));
    const float e0 = expf(w0 - wm);
    const float e1 = expf(w1 - wm);
    const float e2 = expf(w2 - wm);
    const float winv = 1.0f / (e0 + e1 + e2);

    const float s_val = lds_q[d] * (e0 * winv * lds_ksum[d] +
                                    e1 * winv * lds_ksum[64 + d] +
                                    e2 * winv * lds_ksum[128 + d]);
    lds_s[t] = s_val;
    __syncthreads();

    // softmax over d within each r-group of 64
    const int g0 = r * 64;
    float mx = -INFINITY;
    #pragma unroll 8
    for (int i = 0; i < 64; ++i) mx = fmaxf(mx, lds_s[g0 + i]);
    const float ev = expf(s_val - mx);
    lds_e[t] = ev;
    __syncthreads();
    float sm = 0.f;
    #pragma unroll 8
    for (int i = 0; i < 64; ++i) sm += lds_e[g0 + i];

    out[(size_t)b * KC + t] = ev / sm;
}

extern "C" void kernel_launch(void* const* d_in, const int* in_sizes, int n_in,
                              void* d_out, int out_size, void* d_ws, size_t ws_size,
                              hipStream_t stream) {
    const float* q   = (const float*)d_in[0];   // (B, 64, 64)
    const float* k   = (const float*)d_in[1];   // (B, 200, 192)
    /* d_in[2] = v: unused by the reference */
    const int*   kes = (const int*)d_in[3];     // (B, 1)
    const float* mss = (const float*)d_in[4];   // (3, 3)
    float*       out = (float*)d_out;           // (B, 1, 192)

    const int B = in_sizes[3];                  // kes_length has B elements
    aa_kernel<<<B, 192, 0, stream>>>(q, k, kes, mss, out);
}